// S5_40398462386483
// MI455X (gfx1250) — compile-verified
//
#include <hip/hip_runtime.h>
#include <hip/hip_bf16.h>

typedef float v2f __attribute__((ext_vector_type(2)));
typedef float v8f __attribute__((ext_vector_type(8)));

// Problem constants (from reference setup_inputs)
#define BSZ   8
#define LLEN  4096
#define HDIM  1024
#define PDIM  512
#define MTOT  (BSZ * LLEN)      // 32768
#define KTOT  1024              // = H (gemm1) = 2P (gemm2)
#define NTOT  1024              // = 2P (gemm1) = H (gemm2)
#define NCHUNK 16
#define CLEN   256              // LLEN / NCHUNK

// ---------------------------------------------------------------------------
// Setup: W1[h][p] = Re(B_bar[p][h]), W1[h][P+p] = Im(B_bar[p][h]); lam = Lambda_bar
// ---------------------------------------------------------------------------
__global__ __launch_bounds__(256) void k_setup_w1(
    const float* __restrict__ Lre, const float* __restrict__ Lim,
    const float* __restrict__ logstep, const float* __restrict__ Bp,
    float* __restrict__ W1, float* __restrict__ lam)
{
    int idx = blockIdx.x * 256 + threadIdx.x;   // P*H threads
    int p = idx >> 10;           // / HDIM
    int h = idx & (HDIM - 1);
    float lr = fminf(Lre[p], -1e-4f);
    float li = Lim[p];
    float dt = __expf(logstep[p]);
    float er = __expf(lr * dt);
    float lbr = er * __cosf(li * dt);
    float lbi = er * __sinf(li * dt);
    // coef = (Lambda_bar - 1) / Lambda
    float nr = lbr - 1.0f, ni = lbi;
    float den = 1.0f / (lr * lr + li * li);
    float cr = (nr * lr + ni * li) * den;
    float ci = (ni * lr - nr * li) * den;
    float br = Bp[((size_t)p * HDIM + h) * 2 + 0];
    float bi = Bp[((size_t)p * HDIM + h) * 2 + 1];
    W1[(size_t)h * NTOT + p]        = cr * br - ci * bi;   // real col
    W1[(size_t)h * NTOT + PDIM + p] = cr * bi + ci * br;   // imag col
    if (h == 0) { lam[p] = lbr; lam[PDIM + p] = lbi; }
}

// W2[p][h] = Re(C[h][p]),  W2[P+p][h] = -Im(C[h][p])
__global__ __launch_bounds__(256) void k_setup_w2(
    const float* __restrict__ Cp, float* __restrict__ W2)
{
    int idx = blockIdx.x * 256 + threadIdx.x;   // P*H threads
    int p = idx >> 10;
    int h = idx & (HDIM - 1);
    float cr = Cp[((size_t)h * PDIM + p) * 2 + 0];
    float ci = Cp[((size_t)h * PDIM + p) * 2 + 1];
    W2[(size_t)p * HDIM + h]          =  cr;
    W2[(size_t)(PDIM + p) * HDIM + h] = -ci;
}

// ---------------------------------------------------------------------------
// fp32 WMMA GEMM: C = A(MxK) * B(KxN), optional epilogue C += D[n]*U[m][n]
// Block tile 128x128, 8 waves (4M x 2N) of 32x64, K staged by 16 through LDS.
// Both As and Bs stored fragment-major (padded stride 20) so A and B frags
// are single 8B-aligned ds_load_2addr_b64 reads. Global loads for the next
// K-stage are issued before the WMMA compute on the current stage.
// ---------------------------------------------------------------------------
template <int ADD_DU>
__global__ __launch_bounds__(256) void k_gemm_wmma(
    const float* __restrict__ A, const float* __restrict__ Bm,
    float* __restrict__ C, const float* __restrict__ Uepi,
    const float* __restrict__ Depi)
{
    __shared__ float As[128][20];   // [m][k], stride 20: aligned + bank-spread
    __shared__ float Bs[128][20];   // [n][k] (transposed tile), same layout

    const int tid  = threadIdx.x;
    const int lane = tid & 31;
    const int w    = tid >> 5;           // 0..7
    const int half = lane >> 4;          // 0/1
    const int lm   = lane & 15;
    const int wm   = (w >> 2) * 32;      // wave M offset in block (2 rows of waves)... see below
    // waves arranged 4 (M) x 2 (N): wave tile 32 (M) x 64 (N)
    const int wmo  = (w >> 1) * 32;      // M offset: waves 0,1->0 .. 6,7->96? no:
    const int wno  = (w & 1) * 64;       // N offset
    (void)wm;
    const int mBase = blockIdx.y * 128;
    const int nBase = blockIdx.x * 128;

    v8f acc[2][4] = {};                  // 2 M-tiles x 4 N-tiles of 16x16

    const int ar = tid >> 1;             // A-stage row 0..127
    const int ac = (tid & 1) * 8;        // A-stage col base
    const int kb = tid >> 4;             // B-stage k row 0..15
    const int nb = (tid & 15) * 8;       // B-stage n base

    float4 ra0, ra1, rb0, rb1;
    auto load_g = [&](int kk) {
        const float4* ga = (const float4*)(A + (size_t)(mBase + ar) * KTOT + kk + ac);
        ra0 = ga[0];
        ra1 = ga[1];
        const float4* gb = (const float4*)(Bm + (size_t)(kk + kb) * NTOT + nBase + nb);
        rb0 = gb[0];
        rb1 = gb[1];
    };

    load_g(0);
    for (int kk = 0; kk < KTOT; kk += 16) {
        __syncthreads();                 // previous compute done with LDS
        *(float4*)&As[ar][ac]     = ra0;
        *(float4*)&As[ar][ac + 4] = ra1;
        Bs[nb + 0][kb] = rb0.x;          // transpose B tile into [n][k]
        Bs[nb + 1][kb] = rb0.y;
        Bs[nb + 2][kb] = rb0.z;
        Bs[nb + 3][kb] = rb0.w;
        Bs[nb + 4][kb] = rb1.x;
        Bs[nb + 5][kb] = rb1.y;
        Bs[nb + 6][kb] = rb1.z;
        Bs[nb + 7][kb] = rb1.w;
        __syncthreads();
        if (kk + 16 < KTOT) load_g(kk + 16);   // prefetch next stage (hidden by WMMAs)

        #pragma unroll
        for (int k4 = 0; k4 < 16; k4 += 4) {
            const int kA = k4 + 2 * half;  // lanes 0-15: K={k4,k4+1}; 16-31: K={k4+2,k4+3}
            v2f afrag[2], bfrag[4];
            afrag[0] = *(const v2f*)&As[wmo + lm][kA];
            afrag[1] = *(const v2f*)&As[wmo + 16 + lm][kA];
            #pragma unroll
            for (int tn = 0; tn < 4; ++tn)
                bfrag[tn] = *(const v2f*)&Bs[wno + tn * 16 + lm][kA];
            #pragma unroll
            for (int tm = 0; tm < 2; ++tm)
                #pragma unroll
                for (int tn = 0; tn < 4; ++tn)
                    acc[tm][tn] = __builtin_amdgcn_wmma_f32_16x16x4_f32(
                        false, afrag[tm], false, bfrag[tn],
                        (short)0, acc[tm][tn], false, false);
        }
    }

    // Epilogue: C/D layout — VGPR r: lanes 0-15 -> M=r, lanes 16-31 -> M=r+8; N=lane&15
    #pragma unroll
    for (int tm = 0; tm < 2; ++tm) {
        #pragma unroll
        for (int tn = 0; tn < 4; ++tn) {
            const int n = nBase + wno + tn * 16 + lm;
            float dcoef = 0.0f;
            if (ADD_DU) dcoef = Depi[n];
            #pragma unroll
            for (int r = 0; r < 8; ++r) {
                const int m = mBase + wmo + tm * 16 + r + 8 * half;
                const size_t off = (size_t)m * NTOT + n;
                float v = acc[tm][tn][r];
                if (ADD_DU) v += dcoef * Uepi[off];
                C[off] = v;
            }
        }
    }
}

// ---------------------------------------------------------------------------
// Chunked scan: x_l = lam * x_{l-1} + Bu_l  (lam constant per p) — 3 passes.
// Bu layout: row m = b*L+l, col p = real, col P+p = imag (in-place -> xs).
// ---------------------------------------------------------------------------
__global__ __launch_bounds__(256) void k_scan1(
    float* __restrict__ Bu, const float* __restrict__ lam, float* __restrict__ carry)
{
    int idx = blockIdx.x * 256 + threadIdx.x;       // B*NCHUNK*P threads
    int p = idx & (PDIM - 1);
    int c = (idx >> 9) & (NCHUNK - 1);
    int b = idx >> 13;
    float lr = lam[p], li = lam[PDIM + p];
    float xr = 0.0f, xi = 0.0f;
    size_t base = ((size_t)(b * LLEN + c * CLEN)) * NTOT + p;
    for (int l = 0; l < CLEN; ++l) {
        size_t o = base + (size_t)l * NTOT;
        float br = Bu[o], bi = Bu[o + PDIM];
        float nxr = fmaf(lr, xr, fmaf(-li, xi, br));
        float nxi = fmaf(lr, xi, fmaf( li, xr, bi));
        xr = nxr; xi = nxi;
        Bu[o] = xr; Bu[o + PDIM] = xi;
    }
    size_t co = ((size_t)(b * NCHUNK + c)) * NTOT + p;
    carry[co] = xr; carry[co + PDIM] = xi;
}

__global__ __launch_bounds__(256) void k_scan2(
    const float* __restrict__ lam, const float* __restrict__ carry,
    float* __restrict__ prefix)
{
    int idx = blockIdx.x * 256 + threadIdx.x;       // B*P threads
    int p = idx & (PDIM - 1);
    int b = idx >> 9;
    float lr = lam[p], li = lam[PDIM + p];
    // lam^CLEN (=lam^256) by 8 squarings
    float pr = lr, pi = li;
    #pragma unroll
    for (int s = 0; s < 8; ++s) {
        float nr = pr * pr - pi * pi;
        float ni = 2.0f * pr * pi;
        pr = nr; pi = ni;
    }
    float hr = 0.0f, hi = 0.0f;
    for (int c = 0; c < NCHUNK; ++c) {
        size_t o = ((size_t)(b * NCHUNK + c)) * NTOT + p;
        prefix[o] = hr; prefix[o + PDIM] = hi;      // exclusive prefix (state entering chunk c)
        float cr = carry[o], ci = carry[o + PDIM];
        float nhr = fmaf(pr, hr, fmaf(-pi, hi, cr));
        float nhi = fmaf(pr, hi, fmaf( pi, hr, ci));
        hr = nhr; hi = nhi;
    }
}

__global__ __launch_bounds__(256) void k_scan3(
    float* __restrict__ Bu, const float* __restrict__ lam,
    const float* __restrict__ prefix)
{
    int idx = blockIdx.x * 256 + threadIdx.x;       // B*NCHUNK*P threads
    int p = idx & (PDIM - 1);
    int c = (idx >> 9) & (NCHUNK - 1);
    int b = idx >> 13;
    float lr = lam[p], li = lam[PDIM + p];
    size_t po = ((size_t)(b * NCHUNK + c)) * NTOT + p;
    float hr = prefix[po], hi = prefix[po + PDIM];
    float pwr = lr, pwi = li;                       // lam^1
    size_t base = ((size_t)(b * LLEN + c * CLEN)) * NTOT + p;
    for (int l = 0; l < CLEN; ++l) {
        size_t o = base + (size_t)l * NTOT;
        float xr = Bu[o], xi = Bu[o + PDIM];
        xr = fmaf(pwr, hr, fmaf(-pwi, hi, xr));     // x += lam^{l+1} * h_in
        xi = fmaf(pwr, hi, fmaf( pwi, hr, xi));
        Bu[o] = xr; Bu[o + PDIM] = xi;
        float nr = fmaf(pwr, lr, -pwi * li);
        float ni = fmaf(pwr, li,  pwi * lr);
        pwr = nr; pwi = ni;
    }
}

// ---------------------------------------------------------------------------
extern "C" void kernel_launch(void* const* d_in, const int* in_sizes, int n_in,
                              void* d_out, int out_size, void* d_ws, size_t ws_size,
                              hipStream_t stream) {
    const float* u       = (const float*)d_in[0];   // (B, L, H)
    const float* Lre     = (const float*)d_in[1];   // (P,)
    const float* Lim     = (const float*)d_in[2];   // (P,)
    const float* Bp      = (const float*)d_in[3];   // (P, H, 2)
    const float* Cp      = (const float*)d_in[4];   // (H, P, 2)
    const float* Dv      = (const float*)d_in[5];   // (H,)
    const float* logstep = (const float*)d_in[6];   // (P,1) flat
    float* out = (float*)d_out;

    float* ws     = (float*)d_ws;
    float* W1     = ws;                              // 1M floats (H x 2P)
    float* W2     = ws + (1u << 20);                 // 1M floats (2P x H)
    float* lam    = ws + (2u << 20);                 // 1024 floats
    float* carry  = lam + 2048;                      // 131072 floats
    float* prefix = carry + 131072;                  // 131072 floats
    float* Bu     = ws + (4u << 20);                 // 32M floats (M x 2P), in-place -> xs

    // setup weights + discretization
    k_setup_w1<<<(PDIM * HDIM) / 256, 256, 0, stream>>>(Lre, Lim, logstep, Bp, W1, lam);
    k_setup_w2<<<(PDIM * HDIM) / 256, 256, 0, stream>>>(Cp, W2);

    dim3 gemm_grid(NTOT / 128, MTOT / 128);
    // GEMM1: Bu = u @ W1
    k_gemm_wmma<0><<<gemm_grid, 256, 0, stream>>>(u, W1, Bu, nullptr, nullptr);

    // chunked associative scan (in place in Bu)
    k_scan1<<<(BSZ * NCHUNK * PDIM) / 256, 256, 0, stream>>>(Bu, lam, carry);
    k_scan2<<<(BSZ * PDIM) / 256, 256, 0, stream>>>(lam, carry, prefix);
    k_scan3<<<(BSZ * NCHUNK * PDIM) / 256, 256, 0, stream>>>(Bu, lam, prefix);

    // GEMM2: ys = xs @ W2 + D*u
    k_gemm_wmma<1><<<gemm_grid, 256, 0, stream>>>(Bu, W2, out, u, Dv);
}